// MessagePassing_25786983645998
// MI455X (gfx1250) — compile-verified
//
#include <hip/hip_runtime.h>
#include <stdint.h>

// ---------------------------------------------------------------------------
// COO SpMM scatter: out[row[e], :] += vals[e] * emb[col[e], :],  D = 64 fp32.
// Memory-bound: ~840 MB of mostly-L2-resident traffic (emb 25.6MB + out
// 25.6MB both fit the 192MB L2), 0.2 GFLOP -> no matmul structure, no WMMA.
// CDNA5 paths used: global_load_async_to_lds_b32 (ASYNCcnt) double-buffered
// edge staging, ds_load broadcast of metadata, global_prefetch, B128 gathers
// batched for MLP, hardware global_atomic_add_f32 scatter.
// ---------------------------------------------------------------------------

#define WAVE_SZ          32
#define WAVES_PER_BLOCK  8
#define BLOCK_SZ         (WAVE_SZ * WAVES_PER_BLOCK)   // 256 threads = 8 waves
#define TILE_EDGES       32                            // edges staged per wave
#define D_FEAT           64
#define BATCH            8   // gathers kept in flight per half-wave

// ---------------------------------------------------------------------------
__global__ void __launch_bounds__(BLOCK_SZ)
zero_out_kernel(float* __restrict__ out, int n) {
    int n4 = n >> 2;
    float4* out4 = (float4*)out;
    const float4 z = {0.f, 0.f, 0.f, 0.f};
    int stride = gridDim.x * blockDim.x;
    for (int i = blockIdx.x * blockDim.x + threadIdx.x; i < n4; i += stride)
        out4[i] = z;
    for (int i = (n4 << 2) + blockIdx.x * blockDim.x + threadIdx.x; i < n;
         i += stride)
        out[i] = 0.f;
}

// ---------------------------------------------------------------------------
// Issue 3 async dword copies (vals/row/col of one edge per lane) into this
// wave's private LDS tile buffer. Tracked on ASYNCcnt (+3 per call).
__device__ __forceinline__ void async_tile_load(const float* vals,
                                                const int* rows,
                                                const int* cols, int e,
                                                int n_edges, uint32_t lds_v,
                                                uint32_t lds_r,
                                                uint32_t lds_c) {
    if (e < n_edges) {
        asm volatile("global_load_async_to_lds_b32 %0, %1, off"
                     :
                     : "v"(lds_v), "v"((uint64_t)(uintptr_t)(vals + e))
                     : "memory");
        asm volatile("global_load_async_to_lds_b32 %0, %1, off"
                     :
                     : "v"(lds_r), "v"((uint64_t)(uintptr_t)(rows + e))
                     : "memory");
        asm volatile("global_load_async_to_lds_b32 %0, %1, off"
                     :
                     : "v"(lds_c), "v"((uint64_t)(uintptr_t)(cols + e))
                     : "memory");
    }
}

// ---------------------------------------------------------------------------
__global__ void __launch_bounds__(BLOCK_SZ)
spmm_coo_scatter(const float* __restrict__ emb, const float* __restrict__ vals,
                 const int* __restrict__ rows, const int* __restrict__ cols,
                 float* __restrict__ out, int n_edges, int n_nodes) {
    // Per-wave double-buffered edge tile: [wave][buf][ {vals|row|col} x 32 ]
    __shared__ uint32_t s_edge[WAVES_PER_BLOCK][2][3 * TILE_EDGES];

    const int lane   = threadIdx.x & (WAVE_SZ - 1);
    const int wave   = threadIdx.x >> 5;
    const int gwave  = blockIdx.x * WAVES_PER_BLOCK + wave;
    const int nwaves = gridDim.x * WAVES_PER_BLOCK;
    const int ntiles = (n_edges + TILE_EDGES - 1) / TILE_EDGES;

    const int sub  = lane & 15;  // feature quad: floats [sub*4, sub*4+4)
    const int half = lane >> 4;  // which edge of the concurrent pair

    // Generic pointers to __shared__ carry the LDS byte offset in the low
    // 32 bits (4GB-aligned shared aperture) — exactly what the async
    // instruction's LDS-address VGPR wants. This ptrtoint also "escapes"
    // s_edge into the asm, so post-wait reads below are forced reloads.
    auto lds_addr = [&](int buf, int slot) -> uint32_t {
        return (uint32_t)(uintptr_t)&s_edge[wave][buf][slot * TILE_EDGES + lane];
    };

    int tile = gwave;
    int buf  = 0;
    if (tile < ntiles)
        async_tile_load(vals, rows, cols, tile * TILE_EDGES + lane, n_edges,
                        lds_addr(0, 0), lds_addr(0, 1), lds_addr(0, 2));

    for (; tile < ntiles; tile += nwaves) {
        const int nxt = tile + nwaves;
        if (nxt < ntiles) {
            // Stage next tile into the other buffer, then wait only for the
            // current tile's 3 async ops (in-order completion per wave).
            async_tile_load(vals, rows, cols, nxt * TILE_EDGES + lane, n_edges,
                            lds_addr(buf ^ 1, 0), lds_addr(buf ^ 1, 1),
                            lds_addr(buf ^ 1, 2));
            asm volatile("s_wait_asynccnt 3" ::: "memory");
        } else {
            asm volatile("s_wait_asynccnt 0" ::: "memory");
        }

        const uint32_t* sv = &s_edge[wave][buf][0];  // non-volatile -> ds_load

        // Warm the gather path for all 32 rows of this tile (256B row = two
        // 128B lines; the two half-waves split the lines across lanes).
        {
            uint32_t c = sv[2 * TILE_EDGES + lane];
            if (c >= (uint32_t)n_nodes) c = 0;
            const char* p = (const char*)(emb + (size_t)c * D_FEAT);
            __builtin_prefetch(p + half * 128, 0, 3);
        }

        const int base_e = tile * TILE_EDGES;
#pragma unroll
        for (int s0 = 0; s0 < TILE_EDGES / 2; s0 += BATCH) {
            float  w[BATCH];
            int    rr[BATCH];
            float4 x[BATCH];

            // Phase 1: metadata (ds_load broadcast) + issue BATCH independent
            // B128 gathers -> up to 8 loads in flight per half-wave.
#pragma unroll
            for (int k = 0; k < BATCH; ++k) {
                const int idx = (s0 + k) * 2 + half;
                union { uint32_t u; float f; } cv;
                cv.u  = sv[idx];
                w[k]  = cv.f;
                rr[k] = (int)sv[TILE_EDGES + idx];
                uint32_t c = sv[2 * TILE_EDGES + idx];
                if (c >= (uint32_t)n_nodes) c = 0;  // stale LDS on tail lanes
                x[k] = *(const float4*)(emb + (size_t)c * D_FEAT + sub * 4);
            }

            // Phase 2: scale + hardware f32 atomic scatter.
#pragma unroll
            for (int k = 0; k < BATCH; ++k) {
                const int idx = (s0 + k) * 2 + half;
                if (base_e + idx < n_edges) {
                    float* o = out + (size_t)rr[k] * D_FEAT + sub * 4;
                    unsafeAtomicAdd(o + 0, x[k].x * w[k]);
                    unsafeAtomicAdd(o + 1, x[k].y * w[k]);
                    unsafeAtomicAdd(o + 2, x[k].z * w[k]);
                    unsafeAtomicAdd(o + 3, x[k].w * w[k]);
                }
            }
        }
        buf ^= 1;
    }
}

// ---------------------------------------------------------------------------
extern "C" void kernel_launch(void* const* d_in, const int* in_sizes, int n_in,
                              void* d_out, int out_size, void* d_ws,
                              size_t ws_size, hipStream_t stream) {
    const float* emb  = (const float*)d_in[0];
    const float* vals = (const float*)d_in[1];
    const int*   rows = (const int*)d_in[2];
    const int*   cols = (const int*)d_in[3];
    float*       out  = (float*)d_out;
    const int n_edges = in_sizes[1];
    const int n_nodes = in_sizes[0] / D_FEAT;

    // 1) zero the accumulator
    {
        int n4 = out_size >> 2;
        int blocks = (n4 + BLOCK_SZ - 1) / BLOCK_SZ;
        if (blocks < 1) blocks = 1;
        if (blocks > 4096) blocks = 4096;
        zero_out_kernel<<<blocks, BLOCK_SZ, 0, stream>>>(out, out_size);
    }

    // 2) scatter-accumulate edges
    {
        const int ntiles = (n_edges + TILE_EDGES - 1) / TILE_EDGES;  // 50000
        int blocks = 2048;  // 16384 waves, ~3 tiles each
        const int max_blocks = (ntiles + WAVES_PER_BLOCK - 1) / WAVES_PER_BLOCK;
        if (blocks > max_blocks) blocks = max_blocks;
        if (blocks < 1) blocks = 1;
        spmm_coo_scatter<<<blocks, BLOCK_SZ, 0, stream>>>(emb, vals, rows, cols,
                                                          out, n_edges, n_nodes);
    }
}